// HeteroGATLayerReal_7318624272782
// MI455X (gfx1250) — compile-verified
//
#include <hip/hip_runtime.h>
#include <hip/hip_bf16.h>
#include <math.h>

// ---------------------------------------------------------------------------
// HeteroGAT layer for MI455X (gfx1250, wave32).
// Dense projections via V_WMMA_F32_16X16X4_F32 (fp32 tensor path).
// Edge softmax + scatter via atomics (bandwidth-bound phase).
// ---------------------------------------------------------------------------

#define HH 8
#define DD 32
#define HD 256          // H*D
#define N_C 10000
#define N_S 1024
#define IN_C 128
#define IN_S 64
#define E_CC 120000
#define E_CS 10000
#define ALPHA_SLOPE 0.2f

typedef float v2f __attribute__((ext_vector_type(2)));
typedef float v8f __attribute__((ext_vector_type(8)));

// ---------------- monotone float <-> uint encoding for atomicMax ------------
__device__ __forceinline__ unsigned fenc(float f) {
  unsigned u = __float_as_uint(f);
  return (u & 0x80000000u) ? ~u : (u | 0x80000000u);
}
__device__ __forceinline__ float fdec(unsigned k) {
  return (k & 0x80000000u) ? __uint_as_float(k ^ 0x80000000u)
                           : __uint_as_float(~k);
}
__device__ __forceinline__ float lrelu(float v) {
  return v > 0.0f ? v : ALPHA_SLOPE * v;
}

// ---------------------------------------------------------------------------
// GEMM: out[M,N] = A[M,K] @ W[K,N] + bias[N]   (M%16==0, N%16==0, K%4==0)
// One wave computes one 16x16 tile with V_WMMA_F32_16X16X4_F32.
// ---------------------------------------------------------------------------
__global__ void wmma_gemm_bias(const float* __restrict__ A,
                               const float* __restrict__ W,
                               const float* __restrict__ bias,
                               float* __restrict__ out,
                               int M, int K, int N) {
  int gwave = (int)((blockIdx.x * blockDim.x + threadIdx.x) >> 5);
  int lane  = threadIdx.x & 31;
  int ntiles = N >> 4;
  int mtiles = M >> 4;
  if (gwave >= mtiles * ntiles) return;      // wave-uniform exit (EXEC all-1s)
  int mt = gwave / ntiles;
  int nt = gwave - mt * ntiles;

  int half = lane >> 4;                      // lane half selects K pair
  int l    = lane & 15;
  const float* arow = A + (size_t)(mt * 16 + l) * K;
  int col = nt * 16 + l;

  v8f c = {0.f, 0.f, 0.f, 0.f, 0.f, 0.f, 0.f, 0.f};
  for (int k0 = 0; k0 < K; k0 += 4) {
    int ka = k0 + 2 * half;
    v2f a, b;
    a.x = arow[ka];
    a.y = arow[ka + 1];
    b.x = W[(size_t)ka * N + col];
    b.y = W[(size_t)(ka + 1) * N + col];
    c = __builtin_amdgcn_wmma_f32_16x16x4_f32(
        /*neg_a=*/false, a, /*neg_b=*/false, b,
        /*c_mod=*/(short)0, c, /*reuse_a=*/false, /*reuse_b=*/false);
  }

  float bv = bias[col];
  int rbase = mt * 16 + half * 8;            // C/D layout: M = v + 8*half
  #pragma unroll
  for (int v = 0; v < 8; v++)
    out[(size_t)(rbase + v) * N + col] = c[v] + bv;
}

// ---------------------------------------------------------------------------
// a[node,h] = sum_d Wh[node,h,d] * avec[h,d]
// ---------------------------------------------------------------------------
__global__ void rowdot_kernel(const float* __restrict__ Wh,
                              const float* __restrict__ avec,
                              float* __restrict__ out, int nnodes) {
  int i = blockIdx.x * blockDim.x + threadIdx.x;  // node*H + h
  if (i >= nnodes * HH) return;
  int h = i & (HH - 1);
  int node = i >> 3;
  const float* p = Wh + (size_t)node * HD + h * DD;
  const float* q = avec + h * DD;
  float s = 0.f;
  #pragma unroll
  for (int k = 0; k < DD; k++) s += p[k] * q[k];
  out[i] = s;
}

// ---------------------------------------------------------------------------
// Edge pass 1: segment max of leaky-relu(a_s[src]+a_d[dst]) into mkey (encoded)
// ---------------------------------------------------------------------------
__global__ void edge_max_kernel(const int* __restrict__ src,
                                const int* __restrict__ dst,
                                const float* __restrict__ as,
                                const float* __restrict__ ad,
                                unsigned* __restrict__ mkey, int E) {
  int i = blockIdx.x * blockDim.x + threadIdx.x;
  if (i >= E * HH) return;
  int e = i >> 3, h = i & (HH - 1);
  int d = dst[e];
  float v = lrelu(as[src[e] * HH + h] + ad[d * HH + h]);
  atomicMax(&mkey[d * HH + h], fenc(v));
}

// ---------------------------------------------------------------------------
// Edge pass 2: ex = exp(e - m[dst]); store per-edge ex, accumulate denominator
// ---------------------------------------------------------------------------
__global__ void edge_sum_kernel(const int* __restrict__ src,
                                const int* __restrict__ dst,
                                const float* __restrict__ as,
                                const float* __restrict__ ad,
                                const unsigned* __restrict__ mkey,
                                float* __restrict__ alpha,
                                float* __restrict__ ssum, int E) {
  int i = blockIdx.x * blockDim.x + threadIdx.x;
  if (i >= E * HH) return;
  int e = i >> 3, h = i & (HH - 1);
  int d = dst[e];
  float v = lrelu(as[src[e] * HH + h] + ad[d * HH + h]);
  float ex = __expf(v - fdec(mkey[d * HH + h]));
  alpha[i] = ex;
  atomicAdd(&ssum[d * HH + h], ex);
}

// ---------------------------------------------------------------------------
// Edge pass 3: out[dst,h,d] += (ex/sum) * Wh[src,h,d]
// one thread per (edge,h,d): lanes cover consecutive d -> coalesced atomics
// ---------------------------------------------------------------------------
__global__ void edge_scatter_kernel(const int* __restrict__ src,
                                    const int* __restrict__ dst,
                                    const float* __restrict__ alpha,
                                    const float* __restrict__ ssum,
                                    const float* __restrict__ Wh,
                                    float* __restrict__ out, int E) {
  int i = blockIdx.x * blockDim.x + threadIdx.x;
  if (i >= E * HD) return;
  int e = i >> 8;            // /256
  int hd = i & (HD - 1);
  int h = hd >> 5;
  int d = dst[e];
  float a = alpha[e * HH + h] / ssum[d * HH + h];
  atomicAdd(&out[(size_t)d * HD + hd], a * Wh[(size_t)src[e] * HD + hd]);
}

// ---------------------------------------------------------------------------
__global__ void copy_kernel(const float* __restrict__ in,
                            float* __restrict__ out, int n) {
  int i = blockIdx.x * blockDim.x + threadIdx.x;
  if (i < n) out[i] = in[i];
}

__global__ void relu_kernel(float* __restrict__ x, int n) {
  int i = blockIdx.x * blockDim.x + threadIdx.x;
  if (i < n) x[i] = fmaxf(x[i], 0.0f);
}

// ---------------------------------------------------------------------------
extern "C" void kernel_launch(void* const* d_in, const int* in_sizes, int n_in,
                              void* d_out, int out_size, void* d_ws,
                              size_t ws_size, hipStream_t stream) {
  const float* feat_c[3] = {(const float*)d_in[0], (const float*)d_in[1],
                            (const float*)d_in[2]};
  const float* feat_s = (const float*)d_in[3];
  const float* W_node = (const float*)d_in[4];
  const float* b_node = (const float*)d_in[5];
  const float* W_cc   = (const float*)d_in[6];
  const float* b_cc   = (const float*)d_in[7];
  const float* W_cs   = (const float*)d_in[8];
  const float* b_cs   = (const float*)d_in[9];
  const float* W_in   = (const float*)d_in[10];
  const float* b_in   = (const float*)d_in[11];
  const float* attn_cc = (const float*)d_in[12];  // [9,2,H,D]
  const float* attn_cs = (const float*)d_in[13];  // [3,2,H,D]
  const int* ecc_src = (const int*)d_in[14];      // [9,E_CC]
  const int* ecc_dst = (const int*)d_in[15];
  const int* ecs_src = (const int*)d_in[16];      // [3,E_CS]
  const int* ecs_dst = (const int*)d_in[17];

  // ---- workspace layout (floats); total ~47.1 MB ----
  float* ws = (float*)d_ws;
  float* Wh_node = ws;                           // 3 * 10000 * 256
  float* Wh_in   = ws + 7680000;                 // 1024 * 256
  float* Wh_buf  = ws + 7942144;                 // 10000 * 256
  float* a_s     = ws + 10502144;                // 10000 * 8
  float* a_d     = ws + 10582144;                // 10000 * 8
  unsigned* mkey = (unsigned*)(ws + 10662144);   // 10000 * 8
  float* ssum    = ws + 10742144;                // 10000 * 8 (adjacent to mkey)
  float* alpha   = ws + 10822144;                // 120000 * 8

  float* out = (float*)d_out;
  float* out_c[3] = {out, out + 2560000, out + 5120000};
  float* out_s = out + 7680000;

  const int TB = 256;
  hipMemsetAsync(d_out, 0, (size_t)out_size * sizeof(float), stream);

  // ---- self projections (dst attention features) ----
  for (int t = 0; t < 3; t++)
    wmma_gemm_bias<<<1250, TB, 0, stream>>>(
        feat_c[t], W_node + (size_t)t * IN_C * HD, b_node + t * HD,
        Wh_node + (size_t)t * N_C * HD, N_C, IN_C, HD);
  wmma_gemm_bias<<<128, TB, 0, stream>>>(feat_s, W_in, b_in, Wh_in,
                                         N_S, IN_S, HD);
  // out_s starts as Wh_in
  copy_kernel<<<(N_S * HD + TB - 1) / TB, TB, 0, stream>>>(Wh_in, out_s,
                                                           N_S * HD);

  // ---- 9 C->C relations, sequential, reusing Wh_buf ----
  for (int r = 0; r < 9; r++) {
    int s = r / 3, d = r % 3;
    wmma_gemm_bias<<<1250, TB, 0, stream>>>(
        feat_c[s], W_cc + (size_t)r * IN_C * HD, b_cc + r * HD, Wh_buf,
        N_C, IN_C, HD);
    rowdot_kernel<<<(N_C * HH + TB - 1) / TB, TB, 0, stream>>>(
        Wh_buf, attn_cc + (size_t)(r * 2 + 0) * HD, a_s, N_C);
    rowdot_kernel<<<(N_C * HH + TB - 1) / TB, TB, 0, stream>>>(
        Wh_node + (size_t)d * N_C * HD, attn_cc + (size_t)(r * 2 + 1) * HD,
        a_d, N_C);
    // zero mkey (encoded -inf == 0) and ssum (contiguous)
    hipMemsetAsync(mkey, 0, (size_t)2 * N_C * HH * sizeof(float), stream);
    const int* es = ecc_src + (size_t)r * E_CC;
    const int* ed = ecc_dst + (size_t)r * E_CC;
    edge_max_kernel<<<(E_CC * HH + TB - 1) / TB, TB, 0, stream>>>(
        es, ed, a_s, a_d, mkey, E_CC);
    edge_sum_kernel<<<(E_CC * HH + TB - 1) / TB, TB, 0, stream>>>(
        es, ed, a_s, a_d, mkey, alpha, ssum, E_CC);
    edge_scatter_kernel<<<(E_CC * HD + TB - 1) / TB, TB, 0, stream>>>(
        es, ed, alpha, ssum, Wh_buf, out_c[d], E_CC);
  }

  // ---- 3 C->state relations ----
  for (int t = 0; t < 3; t++) {
    wmma_gemm_bias<<<1250, TB, 0, stream>>>(
        feat_c[t], W_cs + (size_t)t * IN_C * HD, b_cs + t * HD, Wh_buf,
        N_C, IN_C, HD);
    rowdot_kernel<<<(N_C * HH + TB - 1) / TB, TB, 0, stream>>>(
        Wh_buf, attn_cs + (size_t)(t * 2 + 0) * HD, a_s, N_C);
    rowdot_kernel<<<(N_S * HH + TB - 1) / TB, TB, 0, stream>>>(
        Wh_in, attn_cs + (size_t)(t * 2 + 1) * HD, a_d, N_S);
    hipMemsetAsync(mkey, 0, (size_t)2 * N_C * HH * sizeof(float), stream);
    const int* es = ecs_src + (size_t)t * E_CS;
    const int* ed = ecs_dst + (size_t)t * E_CS;
    edge_max_kernel<<<(E_CS * HH + TB - 1) / TB, TB, 0, stream>>>(
        es, ed, a_s, a_d, mkey, E_CS);
    edge_sum_kernel<<<(E_CS * HH + TB - 1) / TB, TB, 0, stream>>>(
        es, ed, a_s, a_d, mkey, alpha, ssum, E_CS);
    edge_scatter_kernel<<<(E_CS * HD + TB - 1) / TB, TB, 0, stream>>>(
        es, ed, alpha, ssum, Wh_buf, out_s, E_CS);
  }

  // ---- final ReLU over all outputs ----
  relu_kernel<<<(out_size + TB - 1) / TB, TB, 0, stream>>>(out, out_size);
}